// BidirectionalTrustModel_43954695307999
// MI455X (gfx1250) — compile-verified
//
#include <hip/hip_runtime.h>
#include <hip/hip_bf16.h>

// MI455X (gfx1250) — memory-bound streaming kernel.
// Reads only the inputs the reference uses (~124 MB total) once, coalesced:
//   difficulties_obs (T*N f32), inptasksperf (T*N int2), d_pred (N f32).
// Roofline @ 23.3 TB/s  =>  ~5.3 us. Transcendentals use the hardware trans
// pipe (v_exp_f32 / v_log_f32); step constants are compile-time IEEE-exact so
// no runtime divisions remain except the single 1/count (matches reference).
// The 12-bin weighted reduction per row runs on the WMMA f32 16x16x4 path
// (exact fp32), two 16-row tiles per wave32, B = ones => D[m,*] = row sums.

typedef __attribute__((ext_vector_type(2))) float v2f;
typedef __attribute__((ext_vector_type(8))) float v8f;

#define BINS 12

// t*p for step s:  t = (1+exp(beta*(dp-s)))^(-zeta^2),  p = mask/count.
// Hardware path: exp(x) = exp2(x*log2e) -> v_exp_f32; pow(x,y) = exp2(y*log2(x)).
__device__ __forceinline__ float tp_val(float c, float dp, float inv,
                                        float s, float bl2, float negz2) {
    float ex = __builtin_amdgcn_exp2f(bl2 * (dp - s));       // exp(beta*(dp-s))
    float t  = __builtin_amdgcn_exp2f(negz2 * __builtin_amdgcn_logf(1.0f + ex));
    return (s >= c) ? t * inv : 0.0f;                        // t * p[k]
}

__global__ void __launch_bounds__(256)
trust_kernel(const int2* __restrict__ perf,      // inptasksperf as (T*N) int2
             const float* __restrict__ dobs,     // difficulties_obs (T*N)
             const float* __restrict__ dpred,    // difficulties_pred (N)
             const float* __restrict__ betas,
             const float* __restrict__ zetas,
             float* __restrict__ out,
             int N, int T) {
    const int idx  = blockIdx.x * blockDim.x + threadIdx.x;
    const int n    = (idx < N) ? idx : (N - 1);   // clamp: keep EXEC all-ones
    const int lane = threadIdx.x & 31;
    const int wave = threadIdx.x >> 5;

    const float beta  = betas[0];
    const float zeta  = zetas[0];
    const float negz2 = -(zeta * zeta);
    const float bl2   = beta * 1.44269504088896340736f;      // beta * log2(e)

    // ---- capability-edge scan over T observations (coalesced streaming) ----
    float c  = 0.0f;
    float dp = dpred[n];
    for (int t = 0; t < T; ++t) {
        const int base = t * N + n;
        float d  = dobs[base];
        int2 pf  = perf[base];                    // 8B coalesced load
        bool nz  = (pf.x | pf.y) != 0;            // cat in {1,2}
        bool sc  = (pf.y != 0) & (pf.x == 0);     // cat == 1 (success)
        float cm = sc ? fmaxf(c, d) : fminf(c, d);
        c = nz ? cm : c;
    }

    // ---- count of active bins (p normalization); count >= 1 always ----
    int count = 0;
#pragma unroll
    for (int k = 0; k < BINS; ++k)
        count += (((float)k + 0.5f) / 12.0f >= c) ? 1 : 0;   // k const -> folds
    const float inv = 1.0f / (float)count;       // one IEEE div, matches ref p

    // ---- redistribute per-row params for WMMA A-fragment construction ----
    // Tile 0 covers wave rows 0..15, tile 1 covers rows 16..31.
    // A-matrix layout (16x4 f32): lanes 0-15 supply K={0,1}, lanes 16-31 K={2,3}.
    const int  r0 = lane & 15;           // my A row for tile 0
    const int  r1 = 16 + (lane & 15);    // my A row for tile 1
    const bool hi = (lane >= 16);        // which K pair I supply

    float c0  = __shfl(c,   r0, 32), c1  = __shfl(c,   r1, 32);
    float dp0 = __shfl(dp,  r0, 32), dp1 = __shfl(dp,  r1, 32);
    float iv0 = __shfl(inv, r0, 32), iv1 = __shfl(inv, r1, 32);

    // ---- 12-bin weighted sums via V_WMMA_F32_16X16X4_F32, B = ones ----
    const v2f bones = {1.0f, 1.0f};
    v8f acc0 = {0.f, 0.f, 0.f, 0.f, 0.f, 0.f, 0.f, 0.f};
    v8f acc1 = {0.f, 0.f, 0.f, 0.f, 0.f, 0.f, 0.f, 0.f};
#pragma unroll
    for (int i = 0; i < 3; ++i) {        // 3 chunks of K=4 -> 12 bins
        // Compile-time IEEE-exact step constants for this chunk's 4 bins.
        const float sK0 = ((float)(4 * i + 0) + 0.5f) / 12.0f;
        const float sK1 = ((float)(4 * i + 1) + 0.5f) / 12.0f;
        const float sK2 = ((float)(4 * i + 2) + 0.5f) / 12.0f;
        const float sK3 = ((float)(4 * i + 3) + 0.5f) / 12.0f;
        const float sA = hi ? sK2 : sK0; // my first K slot's step
        const float sB = hi ? sK3 : sK1; // my second K slot's step

        v2f a0 = { tp_val(c0, dp0, iv0, sA, bl2, negz2),
                   tp_val(c0, dp0, iv0, sB, bl2, negz2) };
        v2f a1 = { tp_val(c1, dp1, iv1, sA, bl2, negz2),
                   tp_val(c1, dp1, iv1, sB, bl2, negz2) };
        acc0 = __builtin_amdgcn_wmma_f32_16x16x4_f32(
                   false, a0, false, bones, (short)0, acc0, false, false);
        acc1 = __builtin_amdgcn_wmma_f32_16x16x4_f32(
                   false, a1, false, bones, (short)0, acc1, false, false);
    }

    // ---- scatter row sums back to owning lanes via LDS ----
    // D layout (16x16 f32): VGPR j on lanes 0-15 = row j, lanes 16-31 = row j+8.
    __shared__ float red[8 * 32];        // 256 threads = 8 waves
    float* rw = &red[wave * 32];
    if (lane == 0) {
#pragma unroll
        for (int j = 0; j < 8; ++j) { rw[j]      = acc0[j]; rw[16 + j] = acc1[j]; }
    }
    if (lane == 16) {
#pragma unroll
        for (int j = 0; j < 8; ++j) { rw[8 + j]  = acc0[j]; rw[24 + j] = acc1[j]; }
    }
    __syncthreads();
    float trust = rw[lane];

    if (idx < N) out[idx] = trust;
}

extern "C" void kernel_launch(void* const* d_in, const int* in_sizes, int n_in,
                              void* d_out, int out_size, void* d_ws, size_t ws_size,
                              hipStream_t stream) {
    // setup_inputs order:
    // 0 inptasksobs (unused)  1 inptasksperf (int32, T*N*2)  2 inptaskspred (unused)
    // 3 num_obs_tasks (unused; derived)  4 tasksobsids (unused)  5 taskspredids (unused)
    // 6 difficulties_obs (f32, T*N)  7 difficulties_pred (f32, N)  8 betas  9 zetas
    const int2*  perf  = (const int2*)d_in[1];
    const float* dobs  = (const float*)d_in[6];
    const float* dpred = (const float*)d_in[7];
    const float* betas = (const float*)d_in[8];
    const float* zetas = (const float*)d_in[9];
    float* out = (float*)d_out;

    const int N = in_sizes[7];
    const int T = in_sizes[6] / N;

    dim3 block(256);
    dim3 grid((N + 255) / 256);
    trust_kernel<<<grid, block, 0, stream>>>(perf, dobs, dpred, betas, zetas,
                                             out, N, T);
}